// CovarianceEstimator_76965813944520
// MI455X (gfx1250) — compile-verified
//
#include <hip/hip_runtime.h>

// ---------------------------------------------------------------------------
// Per-pixel 16x16 patch covariance via V_WMMA_F32_16X16X4_F32 (gfx1250).
//   R[pix] = (1/25) * Yc * Yc^T,  Yc = patches(16x25) - mean over 25
// One wave32 = one pixel. Gram product is invariant to permutation of the 25
// patch positions across (chunk,K) slots, so interior pixels use an
// x-adjacent slot mapping: 5 x b64 + 3 x b32 loads per lane, no reflect math.
// Border pixels (3%) take a generic reflect path. pix is scalarized so the
// path split is a scalar branch (EXEC stays all-1s around every WMMA).
// ---------------------------------------------------------------------------

typedef __attribute__((ext_vector_type(2))) float v2f;
typedef __attribute__((ext_vector_type(8))) float v8f;

#define K_CH   16
#define H_DIM  256
#define W_DIM  256
#define NPATCH 25

__device__ __forceinline__ v2f ld2(const float* p) {
    v2f r;
    __builtin_memcpy(&r, p, sizeof(float) * 2);   // 4B-aligned pair load
    return r;
}

__global__ __launch_bounds__(256) void cov_wmma_kernel(const float* __restrict__ S,
                                                       float* __restrict__ R,
                                                       int npix) {
    const int lane = threadIdx.x & 31;
    // wave-uniform pixel id -> SGPR (scalar branches, SADDR-form stores)
    const int pix = __builtin_amdgcn_readfirstlane(
        (int)(blockIdx.x * 8) + (int)(threadIdx.x >> 5));
    if (pix >= npix) return;

    const int b  = pix >> 16;        // H*W = 65536 (fixed by reference)
    const int h  = (pix >> 8) & 255;
    const int w  = pix & 255;
    const int p  = lane & 15;        // channel = matrix row M
    const int hi = lane >> 4;        // half-wave = K-halves of each chunk

    const float* __restrict__ Sc =
        S + (size_t)((b * K_CH + p) * (H_DIM * W_DIM));

    // v[2c],v[2c+1] = this lane's (K0,K1) pair of chunk c; v[12] = chunk6.x
    float v[13];
    float sum;

    if (h >= 2 && h <= H_DIM - 3 && w >= 2 && w <= W_DIM - 3) {
        // ---- fast interior path: pure immediate-offset loads -------------
        const float* Sp = Sc + h * W_DIM + w;
        // chunks 0..4 = patch rows dy=-2..2; hi=0 -> cols(-2,-1), hi=1 -> (0,+1)
        const float* pb = Sp + (hi ? 0 : -2);
        v2f p0 = ld2(pb - 2 * W_DIM);
        v2f p1 = ld2(pb -     W_DIM);
        v2f p2 = ld2(pb);
        v2f p3 = ld2(pb +     W_DIM);
        v2f p4 = ld2(pb + 2 * W_DIM);
        // column +2 singles: hi0 -> rows {-2,-1,+2}, hi1 -> rows {0,+1} (+dummy)
        const float* sb = Sp + 2 + (hi ? 2 * W_DIM : 0);
        float u0 = sb[-2 * W_DIM];                       // s(-2) | s(0)
        float u1 = sb[-W_DIM];                           // s(-1) | s(+1)
        const float* s2p = Sp + 2 + (hi ? W_DIM : 2 * W_DIM);
        float u2 = *s2p;                                 // s(+2) | dummy

        v[0]=p0.x; v[1]=p0.y; v[2]=p1.x; v[3]=p1.y; v[4]=p2.x; v[5]=p2.y;
        v[6]=p3.x; v[7]=p3.y; v[8]=p4.x; v[9]=p4.y; v[10]=u0; v[11]=u1; v[12]=u2;
        sum = p0.x + p0.y + p1.x + p1.y + p2.x + p2.y + p3.x + p3.y
            + p4.x + p4.y + u0 + u1 + (hi ? 0.0f : u2);  // 13 | 12 valid values
    } else {
        // ---- border path: generic reflect indexing -----------------------
        sum = 0.0f;
#pragma unroll
        for (int c = 0; c < 7; ++c) {
#pragma unroll
            for (int t = 0; t < 2; ++t) {
                const int j = 4 * c + 2 * hi + t;
                float val = 0.0f;
                if (j < NPATCH) {
                    int y = h + (j / 5) - 2;
                    int x = w + (j % 5) - 2;
                    y = (y < 0) ? -y : ((y >= H_DIM) ? (2 * H_DIM - 2 - y) : y);
                    x = (x < 0) ? -x : ((x >= W_DIM) ? (2 * W_DIM - 2 - x) : x);
                    val = Sc[y * W_DIM + x];
                }
                if (2 * c + t < 13) v[2 * c + t] = val;  // slot 13 is always 0
                sum += val;
            }
        }
    }

    // lane pair (L, L^16) together covers all 25 values of channel p
    sum += __shfl_xor(sum, 16, 32);
    const float mean = sum * (1.0f / NPATCH);

    // Gram accumulation: same regs as A and B computes Yc * Yc^T
    v8f acc = {0.f, 0.f, 0.f, 0.f, 0.f, 0.f, 0.f, 0.f};
#pragma unroll
    for (int c = 0; c < 6; ++c) {
        v2f a;
        a.x = v[2 * c]     - mean;
        a.y = v[2 * c + 1] - mean;
        acc = __builtin_amdgcn_wmma_f32_16x16x4_f32(
            false, a, false, a, (short)0, acc, false, false);
    }
    {   // chunk 6: (real,0) for hi=0, (0,0) for hi=1
        v2f a;
        a.x = hi ? 0.0f : (v[12] - mean);
        a.y = 0.0f;
        acc = __builtin_amdgcn_wmma_f32_16x16x4_f32(
            false, a, false, a, (short)0, acc, false, false);
    }

    // C/D layout: VGPR m, lanes 0-15 -> (M=m,N=lane); lanes 16-31 -> (M=m+8).
    float* __restrict__ out = R + (size_t)pix * (K_CH * K_CH);
#pragma unroll
    for (int m = 0; m < 8; ++m) {
        out[(m + hi * 8) * 16 + p] = acc[m] * (1.0f / NPATCH);
    }
}

extern "C" void kernel_launch(void* const* d_in, const int* in_sizes, int n_in,
                              void* d_out, int out_size, void* d_ws, size_t ws_size,
                              hipStream_t stream) {
    const float* S = (const float*)d_in[0];
    float* R = (float*)d_out;

    const int npix   = in_sizes[0] / K_CH;   // B*H*W
    const int blocks = (npix + 7) / 8;       // 8 pixel-waves per 256-thread block

    hipLaunchKernelGGL(cov_wmma_kernel, dim3(blocks), dim3(256), 0, stream,
                       S, R, npix);
}